// Graph_GRU_84576495993467
// MI455X (gfx1250) — compile-verified
//
#include <hip/hip_runtime.h>
#include <cstdint>
#include <cstddef>

// Problem constants (match reference)
constexpr int GN = 50000;    // nodes
constexpr int GE = 1600000;  // edges
constexpr int GH = 128;      // hidden
constexpr int GLAYERS = 2;   // layers
constexpr size_t NH = (size_t)GN * GH;

typedef __attribute__((ext_vector_type(2))) float v2f;
typedef __attribute__((ext_vector_type(8))) float v8f;

// ---------------------------------------------------------------------------
// fp32 WMMA: D(16x16) += A(16x4) * B(4x16), wave32.
// A lane layout: row = lane&15, k = 2*(lane>>4) + {0,1}
// B lane layout: col = lane&15, k = 2*(lane>>4) + {0,1}
// C/D: vgpr i -> M = i + 8*(lane>>4), N = lane&15
// ---------------------------------------------------------------------------
__device__ __forceinline__ v8f wmma_f32(v2f a, v2f b, v8f c) {
  return __builtin_amdgcn_wmma_f32_16x16x4_f32(
      /*neg_a=*/false, a, /*neg_b=*/false, b,
      /*c_mod=*/(short)0, c, /*reuse_a=*/false, /*reuse_b=*/false);
}

__device__ __forceinline__ float sigmoid_f(float x) {
  return 1.0f / (1.0f + __expf(-x));
}
__device__ __forceinline__ float tanh_f(float x) {
  return 1.0f - 2.0f / (1.0f + __expf(2.0f * x));
}

// Load 16x128 A tile as WMMA fragments, fused with the GIN "x + agg" add.
__device__ __forceinline__ void load_afrag_sum(const float* __restrict__ X,
                                               const float* __restrict__ A,
                                               int row, int kof, v2f* frag) {
  const float* xr = X + (size_t)row * GH;
  const float* ar = A + (size_t)row * GH;
#pragma unroll
  for (int kk = 0; kk < 32; ++kk) {
    const int c = (kk << 2) + kof;
    const float2 xv = *(const float2*)(xr + c);
    const float2 av = *(const float2*)(ar + c);
    v2f f;
    f.x = xv.x + av.x;
    f.y = xv.y + av.y;
    frag[kk] = f;
  }
}

// Stage a 128x16 column slab of W (row-major [H][H]) into LDS, transposed:
// lds[c*132 + k] = W[k][n0+c].  Pad 132 keeps ds reads conflict-light.
__device__ __forceinline__ void stage_slab(const float* __restrict__ Wg, int n0,
                                           float* lds) {
  const int tid = threadIdx.x;
#pragma unroll
  for (int i = 0; i < 8; ++i) {
    const int idx = i * 256 + tid;  // 0..2047
    const int k = idx >> 4;
    const int c = idx & 15;
    lds[c * 132 + k] = Wg[k * GH + n0 + c];  // 64B-contiguous per 16 lanes
  }
}

// ---------------------------------------------------------------------------
// Scatter-add (segment_sum over edges). One wave == one edge row (32 lanes x
// float4 == 512B).  fp32 global atomics.
// ---------------------------------------------------------------------------
__global__ __launch_bounds__(256) void scatter_add_kernel(
    const int* __restrict__ src, const int* __restrict__ dst,
    const float* __restrict__ feat, float* __restrict__ agg) {
  const unsigned tid = blockIdx.x * 256u + threadIdx.x;
  const unsigned e = tid >> 5;  // edge
  if (e >= (unsigned)GE) return;
  const int c4 = (tid & 31) << 2;  // column (x4)
  const int s = src[e];
  const int d = dst[e];
  const float4 val = *(const float4*)(feat + (size_t)s * GH + c4);
  float* out = agg + (size_t)d * GH + c4;
  atomicAdd(out + 0, val.x);
  atomicAdd(out + 1, val.y);
  atomicAdd(out + 2, val.z);
  atomicAdd(out + 3, val.w);
}

// ---------------------------------------------------------------------------
// Fused z/r gate GEMM:  z = sigmoid(s@W0 + t@W1 + b0 + b1)
//                       r = sigmoid(s@W2 + t@W3 + b2 + b3),  v = r*h
// 256 threads = 8 waves, 16 rows/wave -> 128 rows/block.
// ---------------------------------------------------------------------------
__global__ __launch_bounds__(256) void gemm_zr_kernel(
    const float* __restrict__ x, const float* __restrict__ aggx,
    const float* __restrict__ hl, const float* __restrict__ aggh,
    const float* __restrict__ Wl, const float* __restrict__ bl,
    float* __restrict__ zout, float* __restrict__ vout) {
  __shared__ float lds[4 * 16 * 132];
  const int tid = threadIdx.x;
  const int wave = tid >> 5;
  const int lane = tid & 31;
  const int row0 = blockIdx.x * 128 + wave * 16;
  const int kof = (lane >> 4) << 1;   // 0 or 2
  const int nn = lane & 15;           // local column
  const int rowhi = (lane >> 4) << 3; // +8 rows for upper half-wave

  int arow = row0 + nn;               // A-fragment row (clamped, EXEC all-1)
  if (arow > GN - 1) arow = GN - 1;

  v2f fs[32], ft[32];
  load_afrag_sum(x, aggx, arow, kof, fs);   // s = x + agg_x
  load_afrag_sum(hl, aggh, arow, kof, ft);  // t = h + agg_h

  const float* W0 = Wl + 0 * GH * GH;  // xz
  const float* W1 = Wl + 1 * GH * GH;  // hz
  const float* W2 = Wl + 2 * GH * GH;  // xr
  const float* W3 = Wl + 3 * GH * GH;  // hr
  float* L0 = lds + 0 * 16 * 132;
  float* L1 = lds + 1 * 16 * 132;
  float* L2 = lds + 2 * 16 * 132;
  float* L3 = lds + 3 * 16 * 132;

  for (int ct = 0; ct < GH / 16; ++ct) {
    const int n0 = ct * 16;
    __syncthreads();  // previous tile's LDS reads done
    stage_slab(W0, n0, L0);
    stage_slab(W1, n0, L1);
    stage_slab(W2, n0, L2);
    stage_slab(W3, n0, L3);
    __syncthreads();

    const float* p0 = L0 + nn * 132 + kof;
    const float* p1 = L1 + nn * 132 + kof;
    const float* p2 = L2 + nn * 132 + kof;
    const float* p3 = L3 + nn * 132 + kof;

    v8f az0 = {0, 0, 0, 0, 0, 0, 0, 0}, az1 = {0, 0, 0, 0, 0, 0, 0, 0};
    v8f ar0 = {0, 0, 0, 0, 0, 0, 0, 0}, ar1 = {0, 0, 0, 0, 0, 0, 0, 0};
#pragma unroll
    for (int kk = 0; kk < 32; kk += 2) {  // 4 independent wmma chains
      az0 = wmma_f32(fs[kk], *(const v2f*)(p0 + 4 * kk), az0);
      az0 = wmma_f32(ft[kk], *(const v2f*)(p1 + 4 * kk), az0);
      ar0 = wmma_f32(fs[kk], *(const v2f*)(p2 + 4 * kk), ar0);
      ar0 = wmma_f32(ft[kk], *(const v2f*)(p3 + 4 * kk), ar0);
      az1 = wmma_f32(fs[kk + 1], *(const v2f*)(p0 + 4 * kk + 4), az1);
      az1 = wmma_f32(ft[kk + 1], *(const v2f*)(p1 + 4 * kk + 4), az1);
      ar1 = wmma_f32(fs[kk + 1], *(const v2f*)(p2 + 4 * kk + 4), ar1);
      ar1 = wmma_f32(ft[kk + 1], *(const v2f*)(p3 + 4 * kk + 4), ar1);
    }
    const v8f accz = az0 + az1;
    const v8f accr = ar0 + ar1;

    const float bz = bl[0 * GH + n0 + nn] + bl[1 * GH + n0 + nn];
    const float br = bl[2 * GH + n0 + nn] + bl[3 * GH + n0 + nn];
#pragma unroll
    for (int i = 0; i < 8; ++i) {
      const int row = row0 + rowhi + i;
      if (row < GN) {
        const size_t off = (size_t)row * GH + n0 + nn;
        const float zv = sigmoid_f(accz[i] + bz);
        const float rv = sigmoid_f(accr[i] + br);
        zout[off] = zv;
        vout[off] = rv * hl[off];  // v = r * h
      }
    }
  }
}

// ---------------------------------------------------------------------------
// Fused candidate + GRU blend:
//   h~ = tanh(s@W4 + u@W5 + b4 + b5),  out = z*h + (1-z)*h~
// ---------------------------------------------------------------------------
__global__ __launch_bounds__(256) void gemm_ht_kernel(
    const float* __restrict__ x, const float* __restrict__ aggx,
    const float* __restrict__ v, const float* __restrict__ aggv,
    const float* __restrict__ Wl, const float* __restrict__ bl,
    const float* __restrict__ zbuf, const float* __restrict__ hl,
    float* __restrict__ outp) {
  __shared__ float lds[2 * 16 * 132];
  const int tid = threadIdx.x;
  const int wave = tid >> 5;
  const int lane = tid & 31;
  const int row0 = blockIdx.x * 128 + wave * 16;
  const int kof = (lane >> 4) << 1;
  const int nn = lane & 15;
  const int rowhi = (lane >> 4) << 3;

  int arow = row0 + nn;
  if (arow > GN - 1) arow = GN - 1;

  v2f fs[32], fu[32];
  load_afrag_sum(x, aggx, arow, kof, fs);  // s = x + agg_x
  load_afrag_sum(v, aggv, arow, kof, fu);  // u = v + agg_v

  const float* W4 = Wl + 4 * GH * GH;  // xh
  const float* W5 = Wl + 5 * GH * GH;  // hh
  float* L4 = lds + 0 * 16 * 132;
  float* L5 = lds + 1 * 16 * 132;

  for (int ct = 0; ct < GH / 16; ++ct) {
    const int n0 = ct * 16;
    __syncthreads();
    stage_slab(W4, n0, L4);
    stage_slab(W5, n0, L5);
    __syncthreads();

    const float* p4 = L4 + nn * 132 + kof;
    const float* p5 = L5 + nn * 132 + kof;

    v8f a0 = {0, 0, 0, 0, 0, 0, 0, 0}, a1 = {0, 0, 0, 0, 0, 0, 0, 0};
#pragma unroll
    for (int kk = 0; kk < 32; kk += 2) {
      a0 = wmma_f32(fs[kk], *(const v2f*)(p4 + 4 * kk), a0);
      a0 = wmma_f32(fu[kk], *(const v2f*)(p5 + 4 * kk), a0);
      a1 = wmma_f32(fs[kk + 1], *(const v2f*)(p4 + 4 * kk + 4), a1);
      a1 = wmma_f32(fu[kk + 1], *(const v2f*)(p5 + 4 * kk + 4), a1);
    }
    const v8f acc = a0 + a1;

    const float bh = bl[4 * GH + n0 + nn] + bl[5 * GH + n0 + nn];
#pragma unroll
    for (int i = 0; i < 8; ++i) {
      const int row = row0 + rowhi + i;
      if (row < GN) {
        const size_t off = (size_t)row * GH + n0 + nn;
        const float ht = tanh_f(acc[i] + bh);
        const float zv = zbuf[off];
        outp[off] = zv * hl[off] + (1.0f - zv) * ht;
      }
    }
  }
}

// ---------------------------------------------------------------------------
// Host orchestration (graph-capture safe: only kernels + hipMemsetAsync).
// Inputs: 0=inp[N,H] f32, 1=edgidx[2,E] i32, 2=h[L,N,H] f32,
//         3=W[L,6,H,H] f32, 4=b[L,6,H] f32.  Output: h_out[L,N,H] f32.
// Workspace: agg_x | agg_t(/agg_v) | z | v  (4 * N*H floats = ~102 MB)
// ---------------------------------------------------------------------------
extern "C" void kernel_launch(void* const* d_in, const int* in_sizes, int n_in,
                              void* d_out, int out_size, void* d_ws,
                              size_t ws_size, hipStream_t stream) {
  (void)in_sizes; (void)n_in; (void)out_size; (void)ws_size;
  const float* inp = (const float*)d_in[0];
  const int* edg = (const int*)d_in[1];
  const float* hmat = (const float*)d_in[2];
  const float* Wmat = (const float*)d_in[3];
  const float* bmat = (const float*)d_in[4];
  float* out = (float*)d_out;

  const int* src = edg;       // edgidx[0, :]
  const int* dst = edg + GE;  // edgidx[1, :]

  float* agg_x = (float*)d_ws;
  float* agg_t = agg_x + NH;  // reused for agg_v after z/r computed
  float* zbuf = agg_t + NH;
  float* vbuf = zbuf + NH;

  const int scat_blocks = (int)(((size_t)GE * 32 + 255) / 256);
  const int gemm_blocks = (GN + 127) / 128;

  for (int l = 0; l < GLAYERS; ++l) {
    const float* x = (l == 0) ? inp : (out + (size_t)(l - 1) * NH);
    const float* hl = hmat + (size_t)l * NH;
    const float* Wl = Wmat + (size_t)l * 6 * GH * GH;
    const float* bl = bmat + (size_t)l * 6 * GH;
    float* outl = out + (size_t)l * NH;

    hipMemsetAsync(agg_x, 0, NH * sizeof(float), stream);
    hipMemsetAsync(agg_t, 0, NH * sizeof(float), stream);
    scatter_add_kernel<<<scat_blocks, 256, 0, stream>>>(src, dst, x, agg_x);
    scatter_add_kernel<<<scat_blocks, 256, 0, stream>>>(src, dst, hl, agg_t);

    gemm_zr_kernel<<<gemm_blocks, 256, 0, stream>>>(x, agg_x, hl, agg_t, Wl,
                                                    bl, zbuf, vbuf);

    hipMemsetAsync(agg_t, 0, NH * sizeof(float), stream);  // now agg_v
    scatter_add_kernel<<<scat_blocks, 256, 0, stream>>>(src, dst, vbuf, agg_t);

    gemm_ht_kernel<<<gemm_blocks, 256, 0, stream>>>(x, agg_x, vbuf, agg_t, Wl,
                                                    bl, zbuf, hl, outl);
  }
}